// LGMLoss_v0_38113539784899
// MI455X (gfx1250) — compile-verified
//
#include <hip/hip_runtime.h>

typedef __attribute__((ext_vector_type(2))) float v2f;
typedef __attribute__((ext_vector_type(4))) float v4f;
typedef __attribute__((ext_vector_type(8))) float v8f;

#define BATCH       2048
#define NUM_CLASSES 1000
#define NPAD        1024
#define FEAT_DIM    128

// ---------------------------------------------------------------------------
// Kernel 1: row squared-norms of feat (BATCH rows) and centers (NPAD rows,
// rows >= NUM_CLASSES get 0). One thread per row, float4 loads.
// ---------------------------------------------------------------------------
__global__ __launch_bounds__(256)
void lgm_norms_kernel(const float* __restrict__ feat,
                      const float* __restrict__ centers,
                      float* __restrict__ fnorm,
                      float* __restrict__ cnorm) {
    int i = blockIdx.x * blockDim.x + threadIdx.x;
    if (i < BATCH) {
        const v4f* row = (const v4f*)(feat + (size_t)i * FEAT_DIM);
        float acc = 0.f;
        #pragma unroll
        for (int k = 0; k < FEAT_DIM / 4; ++k) {
            v4f v = row[k];
            acc += v.x * v.x + v.y * v.y + v.z * v.z + v.w * v.w;
        }
        fnorm[i] = acc;
    } else if (i < BATCH + NPAD) {
        int c = i - BATCH;
        float acc = 0.f;
        if (c < NUM_CLASSES) {
            const v4f* row = (const v4f*)(centers + (size_t)c * FEAT_DIM);
            #pragma unroll
            for (int k = 0; k < FEAT_DIM / 4; ++k) {
                v4f v = row[k];
                acc += v.x * v.x + v.y * v.y + v.z * v.z + v.w * v.w;
            }
        }
        cnorm[c] = acc;
    }
}

// ---------------------------------------------------------------------------
// Kernel 2: WMMA FP32 GEMM epilogue-fused distance kernel.
// One wave computes a 16(M) x 64(N) strip: 4 N-tiles of 16x16, sharing the A
// fragment across the 4 WMMAs per k-step.
//
// V_WMMA_F32_16X16X4_F32 fragment layout (wave32):
//   A (16x4):  lane l -> {feat[m = l&15][k0 + 2*(l>>4)], feat[m][k0+2*(l>>4)+1]}
//   B (4x16):  lane l -> {centers[n = l&15][k0 + 2*(l>>4)], ... +1}   (B = centers^T)
//   C/D (16x16, 8 VGPRs): VGPR v, lanes 0-15 -> M=v, lanes 16-31 -> M=v+8; N = l&15
//
// k-loop is fully unrolled and software-pipelined (register double buffer):
// loads for step k+4 are live across the WMMAs of step k, preventing the
// allocator from recycling load registers (no per-WMMA s_wait_loadcnt 0).
//
// Out-of-range columns (>= NUM_CLASSES, last N group only) load from a clamped
// row pointer unconditionally; their stores are predicated off in the epilogue.
// ---------------------------------------------------------------------------
__global__ __launch_bounds__(32)
void lgm_gemm_kernel(const float* __restrict__ feat,
                     const float* __restrict__ centers,
                     const int*   __restrict__ label,
                     const float* __restrict__ fnorm,
                     const float* __restrict__ cnorm,
                     float* __restrict__ logits,
                     float* __restrict__ margin_logits) {
    const int lane   = threadIdx.x;      // 0..31
    const int hl     = lane >> 4;        // 0/1 half-wave
    const int l15    = lane & 15;
    const int tileM  = blockIdx.y;       // 0..127
    const int tileN0 = blockIdx.x * 4;   // 4 consecutive 16-wide N tiles

    const int mrow = tileM * 16 + l15;   // row this lane loads for the A fragment

    int  colv[4];
    bool cok[4];
    #pragma unroll
    for (int j = 0; j < 4; ++j) {
        colv[j] = (tileN0 + j) * 16 + l15;             // column this lane handles
        cok[j]  = colv[j] < NUM_CLASSES;
    }

    const float* __restrict__ ap  = feat    + (unsigned)mrow * FEAT_DIM + 2u * hl;
    const float* __restrict__ bp0 = centers + (unsigned)(cok[0] ? colv[0] : 0) * FEAT_DIM + 2u * hl;
    const float* __restrict__ bp1 = centers + (unsigned)(cok[1] ? colv[1] : 0) * FEAT_DIM + 2u * hl;
    const float* __restrict__ bp2 = centers + (unsigned)(cok[2] ? colv[2] : 0) * FEAT_DIM + 2u * hl;
    const float* __restrict__ bp3 = centers + (unsigned)(cok[3] ? colv[3] : 0) * FEAT_DIM + 2u * hl;

    v8f acc0 = {}, acc1 = {}, acc2 = {}, acc3 = {};

    // ---- software-pipelined, fully unrolled k loop (prefetch distance 1) ----
    v2f a  = *(const v2f*)(ap);
    v2f b0 = *(const v2f*)(bp0);
    v2f b1 = *(const v2f*)(bp1);
    v2f b2 = *(const v2f*)(bp2);
    v2f b3 = *(const v2f*)(bp3);

    #pragma unroll
    for (int k0 = 0; k0 < FEAT_DIM; k0 += 4) {
        v2f an = {}, bn0 = {}, bn1 = {}, bn2 = {}, bn3 = {};
        if (k0 + 4 < FEAT_DIM) {               // constant-folded per unrolled step
            an  = *(const v2f*)(ap  + k0 + 4);
            bn0 = *(const v2f*)(bp0 + k0 + 4);
            bn1 = *(const v2f*)(bp1 + k0 + 4);
            bn2 = *(const v2f*)(bp2 + k0 + 4);
            bn3 = *(const v2f*)(bp3 + k0 + 4);
        }
        acc0 = __builtin_amdgcn_wmma_f32_16x16x4_f32(false, a, false, b0, (short)0, acc0, false, false);
        acc1 = __builtin_amdgcn_wmma_f32_16x16x4_f32(false, a, false, b1, (short)0, acc1, false, false);
        acc2 = __builtin_amdgcn_wmma_f32_16x16x4_f32(false, a, false, b2, (short)0, acc2, false, false);
        acc3 = __builtin_amdgcn_wmma_f32_16x16x4_f32(false, a, false, b3, (short)0, acc3, false, false);
        a = an; b0 = bn0; b1 = bn1; b2 = bn2; b3 = bn3;
    }

    // Per-lane M rows for the C/D layout: m = tileM*16 + v + 8*hl, v = 0..7
    const int mbase = tileM * 16 + 8 * hl;
    float fn[8];
    int   lb[8];
    #pragma unroll
    for (int v = 0; v < 8; ++v) {
        fn[v] = fnorm[mbase + v];
        lb[v] = label[mbase + v];
    }

    v8f accs[4] = {acc0, acc1, acc2, acc3};
    #pragma unroll
    for (int j = 0; j < 4; ++j) {
        const int   col = colv[j];
        const float cn  = cnorm[col];      // always in-bounds: col < NPAD
        #pragma unroll
        for (int v = 0; v < 8; ++v) {
            const int   m     = mbase + v;
            const float dist  = fn[v] + cn - 2.0f * accs[j][v];
            const float logit = -0.5f * dist;
            const float marg  = (lb[v] == col) ? 2.0f * logit : logit; // ALPHA = 1
            const size_t idx  = (size_t)m * NUM_CLASSES + col;
            if (cok[j]) {
                logits[idx]        = logit;
                margin_logits[idx] = marg;
            }
        }
    }
}

// ---------------------------------------------------------------------------
// Kernel 3: likelihood = (1/B) * sum_b ||feat_b - centers_{label_b}||^2 / 2.
// Single 1024-thread block, fixed-order LDS tree reduction -> deterministic.
// ---------------------------------------------------------------------------
__global__ __launch_bounds__(1024)
void lgm_lik_kernel(const float* __restrict__ feat,
                    const float* __restrict__ centers,
                    const int*   __restrict__ label,
                    float* __restrict__ out_lik) {
    __shared__ float red[1024];
    const int t = threadIdx.x;
    float acc = 0.f;
    const int nvec = BATCH * (FEAT_DIM / 4);           // float4 elements
    for (int i = t; i < nvec; i += 1024) {
        const int b  = i >> 5;                         // 128/4 = 32 float4 per row
        const int d4 = i & 31;
        const int c  = label[b];
        v4f f = *(const v4f*)(feat    + (size_t)b * FEAT_DIM + 4 * d4);
        v4f g = *(const v4f*)(centers + (size_t)c * FEAT_DIM + 4 * d4);
        const float dx = f.x - g.x, dy = f.y - g.y, dz = f.z - g.z, dw = f.w - g.w;
        acc += dx * dx + dy * dy + dz * dz + dw * dw;
    }
    red[t] = acc;
    __syncthreads();
    #pragma unroll
    for (int s = 512; s > 0; s >>= 1) {
        if (t < s) red[t] += red[t + s];
        __syncthreads();
    }
    if (t == 0) out_lik[0] = red[0] * (1.0f / (2.0f * BATCH));
}

// ---------------------------------------------------------------------------
extern "C" void kernel_launch(void* const* d_in, const int* in_sizes, int n_in,
                              void* d_out, int out_size, void* d_ws, size_t ws_size,
                              hipStream_t stream) {
    const float* feat    = (const float*)d_in[0];   // [2048, 128] f32
    const int*   label   = (const int*)  d_in[1];   // [2048] int
    const float* centers = (const float*)d_in[2];   // [1000, 128] f32

    float* out    = (float*)d_out;
    float* logits = out;                                        // [2048,1000]
    float* margin = out + (size_t)BATCH * NUM_CLASSES;          // [2048,1000]
    float* lik    = out + (size_t)2 * BATCH * NUM_CLASSES;      // scalar

    float* fnorm = (float*)d_ws;        // 2048 floats
    float* cnorm = fnorm + BATCH;       // 1024 floats

    // 1) norms
    lgm_norms_kernel<<<(BATCH + NPAD + 255) / 256, 256, 0, stream>>>(
        feat, centers, fnorm, cnorm);

    // 2) fused WMMA distance/logits: grid = (N tile groups of 64, M tiles of 16)
    dim3 grid(NPAD / 64, BATCH / 16);
    lgm_gemm_kernel<<<grid, 32, 0, stream>>>(
        feat, centers, label, fnorm, cnorm, logits, margin);

    // 3) likelihood scalar
    lgm_lik_kernel<<<1, 1024, 0, stream>>>(feat, centers, label, lik);
}